// TGDRP_44908178047358
// MI455X (gfx1250) — compile-verified
//
#include <hip/hip_runtime.h>

// ---------------------------------------------------------------------------
// CDNA5 (gfx1250) forward pass for the TGDRP-style GNN reference.
// Dense GEMMs: V_WMMA_F32_16X16X4_F32, A and B^T fragments via global_load_b64,
// no guards in the hot loop (all shapes padded to 16/4 multiples).
// Edge phases: atomic scatter/gather (memory-bound; fits in 192MB L2).
// ---------------------------------------------------------------------------

typedef __attribute__((ext_vector_type(2))) float v2f;
typedef __attribute__((ext_vector_type(8))) float v8f;

#define DEVFN static __device__ __forceinline__

DEVFN float lrelu(float x) { return x > 0.f ? x : 0.2f * x; }
// order-preserving float->int map (involution) for atomicMax on floats
DEVFN int   fenc(float f) { int i = __float_as_int(f); return i >= 0 ? i : (i ^ 0x7fffffff); }
DEVFN float fdec(int i)   { return __int_as_float(i >= 0 ? i : (i ^ 0x7fffffff)); }

DEVFN v8f wmma4(v2f a, v2f b, v8f c) {
  return __builtin_amdgcn_wmma_f32_16x16x4_f32(false, a, false, b, (short)0, c,
                                               false, false);
}

// ---------------------------------------------------------------------------
// Fast f32 WMMA GEMM:  C[M,N] = act(A[M,K] @ B + bias[N]),  B given as Bt[N,K].
// Preconditions: M%64==0 or M%16==0 with grid sized exactly, N%16==0, K%16==0,
// A/Bt rows 8-byte aligned (K even). One wave32 per 16x16 tile, 4 waves/block
// sharing the N-tile (B fragments hit WGP$). act: 0=none, 1=relu, 2=elu.
// ---------------------------------------------------------------------------
__global__ __launch_bounds__(128) void k_gemm_wmma_nt(
    const float* __restrict__ A, const float* __restrict__ Bt,
    const float* __restrict__ bias, float* __restrict__ C,
    int M, int N, int K, int act) {
  const int lane = threadIdx.x;         // 0..31
  const int m    = lane & 15;
  const int half = lane >> 4;           // selects K pair {2h, 2h+1} within k4
  const int tn   = blockIdx.x * 16;
  const int tm   = (blockIdx.y * 4 + threadIdx.y) * 16;
  if (tm >= M) return;                  // wave-uniform

  const float* ap = A  + (size_t)(tm + m) * K + half * 2;
  const float* bp = Bt + (size_t)(tn + m) * K + half * 2;

  v8f acc = {0.f, 0.f, 0.f, 0.f, 0.f, 0.f, 0.f, 0.f};
  int k = 0;
  for (; k + 16 <= K; k += 16) {
    v2f a0 = *(const v2f*)(ap + k);      v2f b0 = *(const v2f*)(bp + k);
    v2f a1 = *(const v2f*)(ap + k + 4);  v2f b1 = *(const v2f*)(bp + k + 4);
    v2f a2 = *(const v2f*)(ap + k + 8);  v2f b2 = *(const v2f*)(bp + k + 8);
    v2f a3 = *(const v2f*)(ap + k + 12); v2f b3 = *(const v2f*)(bp + k + 12);
    acc = wmma4(a0, b0, acc);
    acc = wmma4(a1, b1, acc);
    acc = wmma4(a2, b2, acc);
    acc = wmma4(a3, b3, acc);
  }
  for (; k < K; k += 4) {                // unused (all K%16==0), kept for safety
    v2f a = *(const v2f*)(ap + k);
    v2f b = *(const v2f*)(bp + k);
    acc = wmma4(a, b, acc);
  }

  const float bz = bias[tn + m];
  for (int v = 0; v < 8; ++v) {
    const int row = tm + v + 8 * half;   // C/D layout: VGPR v -> row v + 8*half
    float val = acc[v] + bz;
    if (act == 1)      val = fmaxf(val, 0.f);
    else if (act == 2) val = val > 0.f ? val : (__expf(val) - 1.f);
    C[(size_t)row * N + tn + m] = val;
  }
}

// ---------------------------------------------------------------------------
// Weight prep: Wt[n*ldt + k] = (k<K && n<N) ? W[k*N + n] : 0   (pad K and N)
// ---------------------------------------------------------------------------
__global__ void k_transpose(const float* __restrict__ W, float* __restrict__ Wt,
                            int K, int N, int ldt, int npad) {
  int i = blockIdx.x * blockDim.x + threadIdx.x;
  if (i >= npad * ldt) return;
  int n = i / ldt, kk = i % ldt;
  Wt[i] = (kk < K && n < N) ? W[(size_t)kk * N + n] : 0.f;
}

__global__ void k_pad_bias(const float* __restrict__ b, float* __restrict__ bp,
                           int n, int npad) {
  int i = blockIdx.x * blockDim.x + threadIdx.x;
  if (i < npad) bp[i] = (i < n) ? b[i] : 0.f;
}

__global__ void k_extract_col0(const float* __restrict__ in, float* __restrict__ out,
                               int rows, int ld) {
  int i = blockIdx.x * blockDim.x + threadIdx.x;
  if (i < rows) out[i] = in[(size_t)i * ld];
}

// ---------------------------------------------------------------------------
// Elementwise / reduction helpers
// ---------------------------------------------------------------------------
__global__ void k_copy_pad_rows(const float* __restrict__ x, float* __restrict__ y,
                                int rows, int cin, int ldo) {
  int i = blockIdx.x * blockDim.x + threadIdx.x;
  if (i >= rows * ldo) return;
  int r = i / ldo, c = i % ldo;
  y[i] = (c < cin) ? x[(size_t)r * cin + c] : 0.f;
}

__global__ void k_concat(const float* __restrict__ a, const float* __restrict__ b,
                         float* __restrict__ z, int rows, int ca, int cb) {
  int i = blockIdx.x * blockDim.x + threadIdx.x;
  int n = rows * (ca + cb);
  if (i >= n) return;
  int r = i / (ca + cb), c = i % (ca + cb);
  z[i] = (c < ca) ? a[r * ca + c] : b[r * cb + (c - ca)];
}

// BatchNorm stats: one block per column; population var; rstd = 1/sqrt(var+eps)
__global__ void k_bn_stats(const float* __restrict__ x, int rows, int cols,
                           float* __restrict__ mean, float* __restrict__ rstd) {
  __shared__ float sh[256], sh2[256];
  const int c = blockIdx.x;
  float s = 0.f, s2 = 0.f;
  for (int r = threadIdx.x; r < rows; r += blockDim.x) {
    float v = x[(size_t)r * cols + c];
    s += v; s2 += v * v;
  }
  sh[threadIdx.x] = s; sh2[threadIdx.x] = s2;
  __syncthreads();
  for (int st = 128; st > 0; st >>= 1) {
    if (threadIdx.x < st) { sh[threadIdx.x] += sh[threadIdx.x + st]; sh2[threadIdx.x] += sh2[threadIdx.x + st]; }
    __syncthreads();
  }
  if (threadIdx.x == 0) {
    float mu  = sh[0] / (float)rows;
    float var = sh2[0] / (float)rows - mu * mu;
    mean[c] = mu;
    rstd[c] = rsqrtf(fmaxf(var, 0.f) + 1e-5f);
  }
}

__global__ void k_bn_apply(float* __restrict__ x, int rows, int cols,
                           const float* __restrict__ mean, const float* __restrict__ rstd,
                           const float* __restrict__ w, const float* __restrict__ b,
                           int affine) {
  int i = blockIdx.x * blockDim.x + threadIdx.x;
  int n = rows * cols;
  if (i >= n) return;
  int c = i % cols;
  float v = (x[i] - mean[c]) * rstd[c];
  if (affine) v = v * w[c] + b[c];
  x[i] = v;
}

// ---------------------------------------------------------------------------
// Drug branch (GIN)
// ---------------------------------------------------------------------------
__global__ void k_gin_scatter(const int* __restrict__ src, const int* __restrict__ dst,
                              const float* __restrict__ x, float* __restrict__ agg,
                              int E, int F, int ldx, int lda) {
  int i = blockIdx.x * blockDim.x + threadIdx.x;
  if (i >= E * F) return;
  int e = i / F, f = i % F;
  atomicAdd(&agg[(size_t)dst[e] * lda + f], x[(size_t)src[e] * ldx + f]);
}

// segment_max over contiguous 32-node groups, 3 feature blocks of 128 cols
__global__ void k_drug_pool(const float* __restrict__ f0, const float* __restrict__ f1,
                            const float* __restrict__ f2, float* __restrict__ xd0) {
  int i = blockIdx.x * blockDim.x + threadIdx.x;
  if (i >= 128 * 384) return;
  int bb = i / 384, c = i % 384;
  const float* s = (c < 128) ? f0 : ((c < 256) ? f1 : f2);
  int cc = c & 127;
  float mx = -3.4e38f;
  for (int r = 0; r < 32; ++r) mx = fmaxf(mx, s[(size_t)(bb * 32 + r) * 128 + cc]);
  xd0[i] = mx;
}

// ---------------------------------------------------------------------------
// Cell branch (GAT, DIM_CELL = 8)
// ---------------------------------------------------------------------------
__global__ void k_gat_transform(const float* __restrict__ x, const float* __restrict__ W,
                                const float* __restrict__ as_, const float* __restrict__ ad_,
                                float* __restrict__ h, float* __restrict__ sl,
                                float* __restrict__ dl, int n, int fin) {
  int i = blockIdx.x * blockDim.x + threadIdx.x;
  if (i >= n) return;
  float hv[8];
  for (int o = 0; o < 8; ++o) {
    float acc = 0.f;
    for (int f = 0; f < fin; ++f) acc += x[(size_t)i * fin + f] * W[f * 8 + o];
    hv[o] = acc;
  }
  float s = 0.f, d = 0.f;
  for (int o = 0; o < 8; ++o) {
    h[(size_t)i * 8 + o] = hv[o];
    s += hv[o] * as_[o];
    d += hv[o] * ad_[o];
  }
  sl[i] = s; dl[i] = d;
}

__global__ void k_gat_init_max(const float* __restrict__ sl, const float* __restrict__ dl,
                               int* __restrict__ menc, int n) {
  int i = blockIdx.x * blockDim.x + threadIdx.x;
  if (i >= n) return;
  menc[i] = fenc(lrelu(sl[i] + dl[i]));   // self-loop logit seeds the max
}

__global__ void k_gat_edge_max(const int* __restrict__ src, const int* __restrict__ dst,
                               const float* __restrict__ sl, const float* __restrict__ dl,
                               int* __restrict__ menc, int E) {
  int e = blockIdx.x * blockDim.x + threadIdx.x;
  if (e >= E) return;
  int s = src[e], d = dst[e];
  atomicMax(&menc[d], fenc(lrelu(sl[s] + dl[d])));
}

__global__ void k_gat_node_sum(const float* __restrict__ sl, const float* __restrict__ dl,
                               const int* __restrict__ menc, float* __restrict__ mf,
                               float* __restrict__ ssum, int n) {
  int i = blockIdx.x * blockDim.x + threadIdx.x;
  if (i >= n) return;
  float m = fdec(menc[i]);
  mf[i] = m;
  ssum[i] = __expf(lrelu(sl[i] + dl[i]) - m);   // self-loop term
}

__global__ void k_gat_edge_sum(const int* __restrict__ src, const int* __restrict__ dst,
                               const float* __restrict__ sl, const float* __restrict__ dl,
                               const float* __restrict__ mf, float* __restrict__ ssum, int E) {
  int e = blockIdx.x * blockDim.x + threadIdx.x;
  if (e >= E) return;
  int s = src[e], d = dst[e];
  atomicAdd(&ssum[d], __expf(lrelu(sl[s] + dl[d]) - mf[d]));
}

__global__ void k_gat_out_init(const float* __restrict__ sl, const float* __restrict__ dl,
                               const float* __restrict__ mf, const float* __restrict__ ssum,
                               const float* __restrict__ h, float* __restrict__ out, int n) {
  int i = blockIdx.x * blockDim.x + threadIdx.x;
  if (i >= n) return;
  float a = __expf(lrelu(sl[i] + dl[i]) - mf[i]) / ssum[i];
  for (int o = 0; o < 8; ++o) out[(size_t)i * 8 + o] = a * h[(size_t)i * 8 + o];
}

__global__ void k_gat_edge_scatter(const int* __restrict__ src, const int* __restrict__ dst,
                                   const float* __restrict__ sl, const float* __restrict__ dl,
                                   const float* __restrict__ mf, const float* __restrict__ ssum,
                                   const float* __restrict__ h, float* __restrict__ out, int E) {
  int e = blockIdx.x * blockDim.x + threadIdx.x;
  if (e >= E) return;
  int s = src[e], d = dst[e];
  float a = __expf(lrelu(sl[s] + dl[d]) - mf[d]) / ssum[d];
  for (int o = 0; o < 8; ++o) atomicAdd(&out[(size_t)d * 8 + o], a * h[(size_t)s * 8 + o]);
}

__global__ void k_bias_relu8(float* __restrict__ x, const float* __restrict__ b, int n8) {
  int i = blockIdx.x * blockDim.x + threadIdx.x;
  if (i >= n8) return;
  x[i] = fmaxf(x[i] + b[i & 7], 0.f);
}

// graclus cluster is globally i//2 -> pairwise row max
__global__ void k_pool_pair(const float* __restrict__ in, float* __restrict__ out, int nout8) {
  int i = blockIdx.x * blockDim.x + threadIdx.x;
  if (i >= nout8) return;
  int r = i >> 3, o = i & 7;
  out[i] = fmaxf(in[(size_t)(2 * r) * 8 + o], in[(size_t)(2 * r + 1) * 8 + o]);
}

// ---------------------------------------------------------------------------
// Host orchestration
// ---------------------------------------------------------------------------
extern "C" void kernel_launch(void* const* d_in, const int* in_sizes, int n_in,
                              void* d_out, int out_size, void* d_ws, size_t ws_size,
                              hipStream_t stream) {
  (void)n_in; (void)out_size; (void)ws_size;

  // ---- constants (reference) ----
  const int Bb = 128, ND = 32, NDRUG = Bb * ND;          // 4096
  const int IN_DRUG = 77, INP = 80, DD = 128;            // K padded 77 -> 80
  const int N0 = 3072, N1 = 1536, N2 = 768, N3 = 384;
  const int NODES[4] = {Bb * N0, Bb * N1, Bb * N2, Bb * N3};

  // ---- inputs (setup_inputs dict order) ----
  const float* cell_x = (const float*)d_in[0];
  const float* drug_x = (const float*)d_in[1];
  const int*   ce[3]  = {(const int*)d_in[2], (const int*)d_in[3], (const int*)d_in[4]};
  const int    Ecell[3] = {in_sizes[2] / 2, in_sizes[3] / 2, in_sizes[4] / 2};
  const int*   de = (const int*)d_in[8];
  const int    Ed = in_sizes[8] / 2;

  const float *gat_W[3], *gat_b[3], *gat_as[3], *gat_ad[3];
  for (int l = 0; l < 3; ++l) {
    gat_W[l]  = (const float*)d_in[10 + 4 * l];
    gat_b[l]  = (const float*)d_in[11 + 4 * l];
    gat_as[l] = (const float*)d_in[12 + 4 * l];
    gat_ad[l] = (const float*)d_in[13 + 4 * l];
  }
  const float *gin_W1[3], *gin_b1[3], *gin_W2[3], *gin_b2[3], *bn_w[3], *bn_b[3];
  for (int l = 0; l < 3; ++l) {
    gin_W1[l] = (const float*)d_in[22 + 6 * l];
    gin_b1[l] = (const float*)d_in[23 + 6 * l];
    gin_W2[l] = (const float*)d_in[24 + 6 * l];
    gin_b2[l] = (const float*)d_in[25 + 6 * l];
    bn_w[l]   = (const float*)d_in[26 + 6 * l];
    bn_b[l]   = (const float*)d_in[27 + 6 * l];
  }
  const float* demb_W  = (const float*)d_in[40];
  const float* demb_b  = (const float*)d_in[41];
  const float* cemb_W1 = (const float*)d_in[42];
  const float* cemb_b1 = (const float*)d_in[43];
  const float* cemb_W2 = (const float*)d_in[44];
  const float* cemb_b2 = (const float*)d_in[45];
  const float* reg_W1  = (const float*)d_in[46];
  const float* reg_b1  = (const float*)d_in[47];
  const float* reg_W2  = (const float*)d_in[48];
  const float* reg_b2  = (const float*)d_in[49];
  const float* reg_W3  = (const float*)d_in[50];
  const float* reg_b3  = (const float*)d_in[51];

  // ---- workspace carve-out (all f32; every count even -> 8B alignment) ----
  float* Wp = (float*)d_ws;
  size_t off = 0;
  auto alloc = [&](size_t n) { float* p = Wp + off; off += n; return p; };
  // activations
  float* agg   = alloc((size_t)NDRUG * DD);   // layer0 uses [4096,80] region
  float* t1    = alloc((size_t)NDRUG * DD);
  float* feat[3];
  for (int l = 0; l < 3; ++l) feat[l] = alloc((size_t)NDRUG * DD);
  float* meanv = alloc(DD);
  float* rstdv = alloc(DD);
  float* xd0   = alloc((size_t)Bb * 384);
  float* xd    = alloc((size_t)Bb * 256);
  float* h     = alloc((size_t)NODES[0] * 8);
  float* sl    = alloc((size_t)NODES[0]);
  float* dl    = alloc((size_t)NODES[0]);
  int*   menc  = (int*)alloc((size_t)NODES[0]);
  float* mf    = alloc((size_t)NODES[0]);
  float* ssum  = alloc((size_t)NODES[0]);
  float* gout  = alloc((size_t)NODES[0] * 8);
  float* xc[3];
  for (int l = 0; l < 3; ++l) xc[l] = alloc((size_t)NODES[l + 1] * 8);
  float* xcA   = alloc((size_t)Bb * 1024);
  float* xcB   = alloc((size_t)Bb * 256);
  float* z     = alloc((size_t)Bb * 512);
  float* r1    = alloc((size_t)Bb * 512);
  float* r2    = alloc((size_t)Bb * 512);
  float* r3    = alloc((size_t)Bb * 16);     // reg3 result, N padded to 16
  // transposed (and padded) weights: Wt[N][Kpad]
  float* t_gW1[3], *t_gW2[3];
  t_gW1[0] = alloc((size_t)DD * INP);
  t_gW1[1] = alloc((size_t)DD * DD);
  t_gW1[2] = alloc((size_t)DD * DD);
  for (int l = 0; l < 3; ++l) t_gW2[l] = alloc((size_t)DD * DD);
  float* t_demb  = alloc((size_t)256 * 384);
  float* t_cemb1 = alloc((size_t)1024 * 3072);
  float* t_cemb2 = alloc((size_t)256 * 1024);
  float* t_reg1  = alloc((size_t)512 * 512);
  float* t_reg2  = alloc((size_t)512 * 512);
  float* t_reg3  = alloc((size_t)16 * 512);
  float* b3p     = alloc(16);

  auto blocks = [](long long n) { return (unsigned)((n + 255) / 256); };
  auto gemm = [&](const float* A_, const float* Bt_, const float* b_, float* C_,
                  int M, int N, int K, int act) {
    dim3 g((unsigned)(N / 16), (unsigned)((M + 63) / 64));
    k_gemm_wmma_nt<<<g, dim3(32, 4), 0, stream>>>(A_, Bt_, b_, C_, M, N, K, act);
  };
  auto transpose = [&](const float* W_, float* Wt_, int K, int N, int ldt, int npad) {
    k_transpose<<<blocks((long long)npad * ldt), 256, 0, stream>>>(W_, Wt_, K, N, ldt, npad);
  };

  // =================== Weight prep (transpose + pad) ===================
  transpose(gin_W1[0], t_gW1[0], IN_DRUG, DD, INP, DD);
  transpose(gin_W1[1], t_gW1[1], DD, DD, DD, DD);
  transpose(gin_W1[2], t_gW1[2], DD, DD, DD, DD);
  for (int l = 0; l < 3; ++l) transpose(gin_W2[l], t_gW2[l], DD, DD, DD, DD);
  transpose(demb_W,  t_demb,  384, 256, 384, 256);
  transpose(cemb_W1, t_cemb1, 3072, 1024, 3072, 1024);
  transpose(cemb_W2, t_cemb2, 1024, 256, 1024, 256);
  transpose(reg_W1,  t_reg1,  512, 512, 512, 512);
  transpose(reg_W2,  t_reg2,  512, 512, 512, 512);
  transpose(reg_W3,  t_reg3,  512, 1, 512, 16);      // pad N: 1 -> 16
  k_pad_bias<<<1, 32, 0, stream>>>(reg_b3, b3p, 1, 16);

  // =================== Drug branch: 3x GIN ===================
  const float* xcur = drug_x;
  for (int l = 0; l < 3; ++l) {
    const int Fin = (l == 0) ? IN_DRUG : DD;
    const int lda = (l == 0) ? INP : DD;
    k_copy_pad_rows<<<blocks((long long)NDRUG * lda), 256, 0, stream>>>(
        xcur, agg, NDRUG, Fin, lda);
    k_gin_scatter<<<blocks((long long)Ed * Fin), 256, 0, stream>>>(
        de, de + Ed, xcur, agg, Ed, Fin, Fin, lda);
    gemm(agg, (l == 0) ? t_gW1[0] : t_gW1[l], gin_b1[l], t1, NDRUG, DD, lda, 1);
    gemm(t1, t_gW2[l], gin_b2[l], feat[l], NDRUG, DD, DD, 1);
    k_bn_stats<<<DD, 256, 0, stream>>>(feat[l], NDRUG, DD, meanv, rstdv);
    k_bn_apply<<<blocks((long long)NDRUG * DD), 256, 0, stream>>>(
        feat[l], NDRUG, DD, meanv, rstdv, bn_w[l], bn_b[l], 1);
    xcur = feat[l];
  }
  k_drug_pool<<<blocks(128 * 384), 256, 0, stream>>>(feat[0], feat[1], feat[2], xd0);
  gemm(xd0, t_demb, demb_b, xd, Bb, 256, 384, 1);

  // =================== Cell branch: 3x GAT + pool + BN ===================
  const float* xin = cell_x;
  int fin = 3;
  for (int l = 0; l < 3; ++l) {
    const int nin = NODES[l], nout = NODES[l + 1], E = Ecell[l];
    const int* src = ce[l];
    const int* dst = ce[l] + E;
    k_gat_transform<<<blocks(nin), 256, 0, stream>>>(xin, gat_W[l], gat_as[l], gat_ad[l],
                                                     h, sl, dl, nin, fin);
    k_gat_init_max<<<blocks(nin), 256, 0, stream>>>(sl, dl, menc, nin);
    k_gat_edge_max<<<blocks(E), 256, 0, stream>>>(src, dst, sl, dl, menc, E);
    k_gat_node_sum<<<blocks(nin), 256, 0, stream>>>(sl, dl, menc, mf, ssum, nin);
    k_gat_edge_sum<<<blocks(E), 256, 0, stream>>>(src, dst, sl, dl, mf, ssum, E);
    k_gat_out_init<<<blocks(nin), 256, 0, stream>>>(sl, dl, mf, ssum, h, gout, nin);
    k_gat_edge_scatter<<<blocks(E), 256, 0, stream>>>(src, dst, sl, dl, mf, ssum, h, gout, E);
    k_bias_relu8<<<blocks((long long)nin * 8), 256, 0, stream>>>(gout, gat_b[l], nin * 8);
    k_pool_pair<<<blocks((long long)nout * 8), 256, 0, stream>>>(gout, xc[l], nout * 8);
    k_bn_stats<<<8, 256, 0, stream>>>(xc[l], nout, 8, meanv, rstdv);
    k_bn_apply<<<blocks((long long)nout * 8), 256, 0, stream>>>(
        xc[l], nout, 8, meanv, rstdv, nullptr, nullptr, 0);
    xin = xc[l];
    fin = 8;
  }
  // xc[2] is [B*N3, 8] == row-major [B, 3072]
  gemm(xc[2], t_cemb1, cemb_b1, xcA, Bb, 1024, N3 * 8, 1);
  gemm(xcA, t_cemb2, cemb_b2, xcB, Bb, 256, 1024, 1);

  // =================== Regression head ===================
  k_concat<<<blocks((long long)Bb * 512), 256, 0, stream>>>(xd, xcB, z, Bb, 256, 256);
  gemm(z, t_reg1, reg_b1, r1, Bb, 512, 512, 2);
  gemm(r1, t_reg2, reg_b2, r2, Bb, 512, 512, 2);
  gemm(r2, t_reg3, b3p, r3, Bb, 16, 512, 0);
  k_extract_col0<<<blocks(Bb), 256, 0, stream>>>(r3, (float*)d_out, Bb, 16);
}